// RCModel_90125593739860
// MI455X (gfx1250) — compile-verified
//
#include <hip/hip_runtime.h>
#include <math.h>

typedef float v2f __attribute__((ext_vector_type(2)));
typedef float v4f __attribute__((ext_vector_type(4)));
typedef float v8f __attribute__((ext_vector_type(8)));

// ---------------- problem constants ----------------
constexpr int T_TOTAL = 1000000;
constexpr int NSTEPS  = T_TOTAL - 1;     // 999,999 RK4 steps
constexpr int NDIM    = 12;
constexpr int NSEG    = 1024;                               // parallel segments
constexpr int SEGLEN  = (NSTEPS + NSEG - 1) / NSEG;         // 977
constexpr int LASTLEN = NSTEPS - (NSEG - 1) * SEGLEN;       // 528
constexpr int WAVES   = NSEG / 16;                          // 64 waves, 16 seg-columns each

// ---------------- workspace layout (floats) ----------------
constexpr int OFF_M  = 0;                 // 16x16 step matrix M (row major)
constexpr int OFF_ML = 256;               // M^SEGLEN
constexpr int OFF_A  = 512;               // a vector (16)
constexpr int OFF_B  = 528;               // b vector (16)
constexpr int OFF_W  = 544;               // w vector (16)
constexpr int OFF_XS = 560;               // segment start states, NSEG*16
constexpr int OFF_SP = OFF_XS + NSEG*16;  // segment affine partials, NSEG*16

static __device__ __forceinline__ v8f wmma4(v2f a, v2f b, v8f c) {
  // D = A(16x4) x B(4x16) + C, f32 — V_WMMA_F32_16X16X4_F32
  return __builtin_amdgcn_wmma_f32_16x16x4_f32(false, a, false, b, (short)0, c, false, false);
}
static __device__ __forceinline__ float swap16(float x) {
  return __shfl_xor(x, 16, 32);   // exchange between wave32 halves
}

// ---------------- kernel 0: build M, M^L, a, b, w on device ----------------
__global__ void rc_setup(const float* __restrict__ t_eval, const float* __restrict__ x0,
                         const float* __restrict__ A, const float* __restrict__ Bm,
                         const float* __restrict__ loads_raw, const float* __restrict__ areas,
                         float* __restrict__ ws, float* __restrict__ out) {
  __shared__ float hA[256], M2[256], M3[256], M4[256], Mm[256], R[256], P[256];
  __shared__ float b0s[16], Bqs[16];
  const int t = threadIdx.x;
  const int i = t >> 4, j = t & 15;
  const float h = t_eval[1] - t_eval[0];

  hA[t] = (i < NDIM && j < NDIM) ? h * A[i*NDIM + j] : 0.f;
  if (t < 16) {
    float b0v = (t < NDIM) ? Bm[t*11 + 0] : 0.f;
    b0s[t] = b0v;
    float bq = 0.f;
    if (t < NDIM) {
      for (int r = 0; r < 10; ++r) {
        float sig = 1.f / (1.f + expf(-loads_raw[10 + r]));  // gain row; action==0
        bq += Bm[t*11 + 1 + r] * (sig * 50.f * areas[r]);
      }
    }
    Bqs[t] = bq;
  }
  __syncthreads();
  { float s = 0.f;
    #pragma unroll
    for (int k = 0; k < 16; ++k) s += hA[i*16+k] * hA[k*16+j];
    M2[t] = s; }
  __syncthreads();
  { float s3 = 0.f, s4 = 0.f;
    #pragma unroll
    for (int k = 0; k < 16; ++k) { s3 += M2[i*16+k]*hA[k*16+j]; s4 += M2[i*16+k]*M2[k*16+j]; }
    M3[t] = s3; M4[t] = s4; }
  __syncthreads();
  Mm[t] = ((i==j)?1.f:0.f) + hA[t] + 0.5f*M2[t] + (1.f/6.f)*M3[t] + (1.f/24.f)*M4[t];
  ws[OFF_M + t] = Mm[t];
  if (t < 16) {
    float m1b=0,m2b=0,m3b=0,m1q=0,m2q=0,m3q=0;
    #pragma unroll
    for (int k = 0; k < 16; ++k) {
      float a1 = hA[t*16+k], a2 = M2[t*16+k], a3 = M3[t*16+k];
      m1b += a1*b0s[k]; m2b += a2*b0s[k]; m3b += a3*b0s[k];
      m1q += a1*Bqs[k]; m2q += a2*Bqs[k]; m3q += a3*Bqs[k];
    }
    const float c = h / 6.f;
    ws[OFF_A + t] = c * (3.f*b0s[t] + 2.f*m1b + 0.75f*m2b + 0.25f*m3b); // u_t + u_m/2
    ws[OFF_B + t] = c * (3.f*b0s[t] +     m1b + 0.25f*m2b);             // u_n + u_m/2
    ws[OFF_W + t] = c * (6.f*Bqs[t] + 3.f*m1q +       m2q + 0.25f*m3q);
    float xv = (t < NDIM) ? x0[t] : 0.f;
    ws[OFF_XS + t] = xv;                 // segment 0 start state
    if (t < NDIM) out[t] = x0[t];        // output row 0
  }
  // M^SEGLEN by binary exponentiation (uniform control flow)
  R[t] = (i==j) ? 1.f : 0.f;
  P[t] = Mm[t];
  __syncthreads();
  unsigned e = (unsigned)SEGLEN;
  while (e) {
    if (e & 1u) {
      float s = 0.f;
      #pragma unroll
      for (int k = 0; k < 16; ++k) s += R[i*16+k]*P[k*16+j];
      __syncthreads(); R[t] = s; __syncthreads();
    }
    e >>= 1;
    if (e) {
      float s = 0.f;
      #pragma unroll
      for (int k = 0; k < 16; ++k) s += P[i*16+k]*P[k*16+j];
      __syncthreads(); P[t] = s; __syncthreads();
    }
  }
  ws[OFF_ML + t] = R[t];
}

// SEGLEN recurrence steps + optional state stores, shared by all loop variants.
// X (16 dims x 16 seg-columns) is in WMMA C/D accumulator layout.
template<bool STORE, bool CLAMP>
static __device__ __forceinline__ void seg_loop(
    v8f& X, const v2f Am[4], const float va[8], const float vb[8], const float vw[8],
    const float* __restrict__ Tout, float* __restrict__ out,
    int sstart, int slen, int up) {
  int t0 = sstart; if (CLAMP && t0 > NSTEPS - 1) t0 = NSTEPS - 1;
  float tot = Tout[t0];
  for (int i = 0; i < SEGLEN; ++i) {
    int ti = sstart + i;
    if (CLAMP && ti > NSTEPS - 1) ti = NSTEPS - 1;
    const float ton = Tout[ti + 1];
    if (!CLAMP) __builtin_prefetch(Tout + ti + 257, 0, 1);  // ~1KB ahead on the stream

    // forcing matrix V directly in accumulator layout
    v8f acc;
    #pragma unroll
    for (int r = 0; r < 8; ++r) acc[r] = fmaf(va[r], tot, fmaf(vb[r], ton, vw[r]));

    // C-layout -> B-operand conversion with only 4 cross-half swaps:
    // each half only consumes 4 swapped registers, so pre-select what to send.
    const float z0 = swap16(up ? X[0] : X[2]);  // lower gets upper X0 ; upper gets lower X2
    const float z1 = swap16(up ? X[1] : X[3]);
    const float z2 = swap16(up ? X[4] : X[6]);
    const float z3 = swap16(up ? X[5] : X[7]);
    v2f B0 = { up ? z0   : X[0], up ? z1   : X[1] };   // S rows 0..3
    v2f B1 = { up ? z2   : X[4], up ? z3   : X[5] };   // S rows 4..7
    v2f B2 = { up ? X[2] : z0,   up ? X[3] : z1   };   // S rows 8..11
    v2f B3 = { up ? X[6] : z2,   up ? X[7] : z3   };   // S rows 12..15

    acc = wmma4(Am[0], B0, acc);
    acc = wmma4(Am[1], B1, acc);
    acc = wmma4(Am[2], B2, acc);
    acc = wmma4(Am[3], B3, acc);
    X = acc;
    tot = ton;

    if (STORE) {
      if (!CLAMP || i < slen) {
        float* op = out + (size_t)(sstart + i + 1) * 12;  // rows are 48B -> 16B aligned
        if (up == 0) {
          v4f p0 = { X[0], X[1], X[2], X[3] };
          v4f p1 = { X[4], X[5], X[6], X[7] };
          __builtin_nontemporal_store(p0, (v4f*)op);
          __builtin_nontemporal_store(p1, (v4f*)(op + 4));
        } else {
          v4f p2 = { X[0], X[1], X[2], X[3] };            // dims 8..11
          __builtin_nontemporal_store(p2, (v4f*)(op + 8));
        }
      }
    }
  }
}

// ---------------- phases 1 & 3: 16 segment recurrences per wave via WMMA ----------------
template<bool STORE_STATES>
__global__ void rc_segments(const float* __restrict__ Tout, float* __restrict__ ws,
                            float* __restrict__ out) {
  const int lane = threadIdx.x;
  const int up   = lane >> 4;      // wave half: rows 0-7 vs 8-15
  const int n    = lane & 15;      // segment column
  const int seg  = blockIdx.x * 16 + n;
  const int sstart = seg * SEGLEN;                      // < 2^20, int is fine
  const int slen   = (seg == NSEG - 1) ? LASTLEN : SEGLEN;

  // A operands: M columns 4k..4k+3; lane holds row m=n, K = {2*up, 2*up+1}
  const float* Mw = ws + OFF_M;
  v2f Am[4];
  #pragma unroll
  for (int k = 0; k < 4; ++k) {
    int col = 4*k + 2*up;
    Am[k].x = Mw[n*16 + col];
    Am[k].y = Mw[n*16 + col + 1];
  }
  // per-lane forcing constants for dims d = r + 8*up
  float va[8], vb[8], vw[8];
  #pragma unroll
  for (int r = 0; r < 8; ++r) {
    int d = r + 8*up;
    va[r] = ws[OFF_A + d]; vb[r] = ws[OFF_B + d]; vw[r] = ws[OFF_W + d];
  }

  v8f X;
  if (STORE_STATES) {
    #pragma unroll
    for (int r = 0; r < 8; ++r) X[r] = ws[OFF_XS + seg*16 + r + 8*up];
  } else {
    #pragma unroll
    for (int r = 0; r < 8; ++r) X[r] = 0.f;   // affine partial starts at 0
  }

  if (blockIdx.x != (unsigned)(WAVES - 1)) {
    seg_loop<STORE_STATES, false>(X, Am, va, vb, vw, Tout, out, sstart, slen, up);
  } else {
    seg_loop<STORE_STATES, true >(X, Am, va, vb, vw, Tout, out, sstart, slen, up);
  }

  if (!STORE_STATES) {
    #pragma unroll
    for (int r = 0; r < 8; ++r) ws[OFF_SP + seg*16 + r + 8*up] = X[r];
  }
}

// ---------------- phase 2: sequential chain across segment boundaries ----------------
__global__ void rc_chain(float* __restrict__ ws) {
  const int lane = threadIdx.x;
  const int d = lane & 15;
  const float* ML = ws + OFF_ML;
  float mlrow[16];
  #pragma unroll
  for (int c = 0; c < 16; ++c) mlrow[c] = ML[d*16 + c];
  float x = ws[OFF_XS + d];
  for (int b = 0; b < NSEG - 1; ++b) {
    float xn = ws[OFF_SP + b*16 + d];
    #pragma unroll
    for (int c = 0; c < 16; ++c) xn = fmaf(mlrow[c], __shfl(x, c, 32), xn);
    x = xn;
    if (lane < 16) ws[OFF_XS + (b + 1)*16 + lane] = x;
  }
}

extern "C" void kernel_launch(void* const* d_in, const int* in_sizes, int n_in,
                              void* d_out, int out_size, void* d_ws, size_t ws_size,
                              hipStream_t stream) {
  const float* t_eval    = (const float*)d_in[0];
  const float* x0        = (const float*)d_in[1];
  const float* A         = (const float*)d_in[2];
  const float* Bm        = (const float*)d_in[3];
  const float* Tout      = (const float*)d_in[4];
  const float* loads_raw = (const float*)d_in[5];
  const float* areas     = (const float*)d_in[6];
  float* out = (float*)d_out;
  float* ws  = (float*)d_ws;

  rc_setup<<<1, 256, 0, stream>>>(t_eval, x0, A, Bm, loads_raw, areas, ws, out);
  rc_segments<false><<<WAVES, 32, 0, stream>>>(Tout, ws, out);  // per-segment partials
  rc_chain<<<1, 32, 0, stream>>>(ws);                           // stitch segment starts
  rc_segments<true><<<WAVES, 32, 0, stream>>>(Tout, ws, out);   // replay + stream states
}